// GINE_layer_42691974922448
// MI455X (gfx1250) — compile-verified
//
#include <hip/hip_runtime.h>

// GINE layer for MI455X (gfx1250), fp32 end-to-end via V_WMMA_F32_16X16X4_F32.
//
// Roofline: ~1.1 GB traffic (edge_feat 205MB + gather ~410MB + scatter ~410MB)
// vs 21.3 GFLOP fp32 -> memory bound at 23.3 TB/s; fp32 WMMA covers the math.
// Weights are pre-packed K-pair-major so every B fragment is ONE contiguous
// 8-byte load straight into an even VGPR pair (no repack movs). A fragments
// are register-resident across all N-tiles.

typedef __attribute__((ext_vector_type(2))) float v2f;
typedef __attribute__((ext_vector_type(8))) float v8f;

#define D_NODE 128
#define D_EDGE 64
#define WAVES  8   // waves per block; 1 wave owns a 16-row tile

static __device__ __forceinline__ v8f wmma_f32(v2f a, v2f b, v8f c) {
  // (neg_a, A, neg_b, B, c_mod, C, reuse_a, reuse_b)
  return __builtin_amdgcn_wmma_f32_16x16x4_f32(false, a, false, b, (short)0, c,
                                               false, false);
}

// ---------------------------------------------------------------------------
// Pack row-major W[rows][cols] into K-pair-major:
//   Wp[((r>>1)*cols + c)*2 + (r&1)] = W[r*cols + c]
// so a B fragment {W[2p][c], W[2p+1][c]} is one contiguous v2f.
// ---------------------------------------------------------------------------
__global__ __launch_bounds__(256) void pack_pairs_kernel(
    const float* __restrict__ W, float* __restrict__ Wp, int rows, int cols) {
  int i = blockIdx.x * 256 + threadIdx.x;
  if (i >= rows * cols) return;
  int r = i / cols, c = i - r * cols;
  Wp[(((r >> 1) * cols + c) << 1) + (r & 1)] = W[i];
}

// ---------------------------------------------------------------------------
// Edge kernel, one wave32 per 16-edge tile:
//   h = relu(ef@W0 + b0)   16x64  (packed W0 in LDS, h staged in LDS)
//   e = relu(h @W1 + b1)   16x128 (packed W1 via L2/WGP$)
//   m = relu(nf[src] + e); atomicAdd(acc[dst], m)
// ---------------------------------------------------------------------------
__global__ __launch_bounds__(256) void gine_edge_kernel(
    const float* __restrict__ ef, const float* __restrict__ nf,
    const float* __restrict__ w0p, const float* __restrict__ b0,
    const float* __restrict__ w1p, const float* __restrict__ b1,
    const int* __restrict__ src, const int* __restrict__ dst,
    float* __restrict__ acc, int n_edges)
{
  __shared__ v2f   sW0[(D_EDGE / 2) * D_EDGE];   // 16 KB, K-pair-major
  __shared__ float sh[WAVES][16 * D_EDGE];       // 32 KB, D->A staging

  const int tid = threadIdx.x;
  {
    const v2f* w0v = (const v2f*)w0p;
    for (int i = tid; i < (D_EDGE / 2) * D_EDGE; i += 256) sW0[i] = w0v[i];
  }

  const int wave  = tid >> 5;
  const int lane  = tid & 31;
  const int row16 = lane & 15;   // M (or N) index within tile
  const int hi    = lane >> 4;   // K-half selector

  const int  tile   = blockIdx.x * WAVES + wave;
  const int  ebase0 = tile * 16;
  const bool active = ebase0 < n_edges;
  const int  ebase  = active ? ebase0 : 0;

  int arow = ebase + row16;
  if (arow >= n_edges) arow = n_edges - 1;
  const v2f* efv   = (const v2f*)ef;          // 32 v2f per edge row
  const int  abase = arow * (D_EDGE / 2);

  // Register-resident A fragments for layer 0 (reused across 4 N-tiles).
  v2f ae[D_EDGE / 4];
  #pragma unroll
  for (int k = 0; k < D_EDGE / 4; ++k) ae[k] = efv[abase + 2 * k + hi];

  // Hoisted gather/scatter element offsets (32-bit, buffers < 2^31 elems).
  int soff[8], doff[8];
  #pragma unroll
  for (int r = 0; r < 8; ++r) {
    int e  = ebase + r + 8 * hi;
    int ec = e < n_edges ? e : n_edges - 1;
    soff[r] = src[ec] * D_NODE + row16;
    doff[r] = dst[ec] * D_NODE + row16;
  }

  __syncthreads();   // sW0 staged

  float* hrow = &sh[wave][0];

  // ---- layer 0: 4 N-tiles x 16 K-steps ----
  #pragma unroll
  for (int nt = 0; nt < D_EDGE / 16; ++nt) {
    v8f c = {};
    #pragma unroll
    for (int k = 0; k < D_EDGE / 4; ++k)
      c = wmma_f32(ae[k], sW0[(2 * k + hi) * D_EDGE + nt * 16 + row16], c);
    const int   col = nt * 16 + row16;
    const float bs  = b0[col];
    #pragma unroll
    for (int r = 0; r < 8; ++r) {
      float v = c[r] + bs;                       // D: row r+8*hi, col
      hrow[(r + 8 * hi) * D_EDGE + col] = v > 0.f ? v : 0.f;
    }
  }
  __syncthreads();   // per-wave h visible across lanes

  // Register-resident A fragments for layer 1.
  const v2f* shv = (const v2f*)hrow;
  v2f ah[D_EDGE / 4];
  #pragma unroll
  for (int k = 0; k < D_EDGE / 4; ++k)
    ah[k] = shv[row16 * (D_EDGE / 2) + 2 * k + hi];

  // ---- layer 1 + gather + relu + scatter: 8 N-tiles x 16 K-steps ----
  const v2f* w1v = (const v2f*)w1p;
  #pragma unroll 1
  for (int nt = 0; nt < D_NODE / 16; ++nt) {
    v8f c = {};
    #pragma unroll
    for (int k = 0; k < D_EDGE / 4; ++k)
      c = wmma_f32(ah[k], w1v[(2 * k + hi) * D_NODE + nt * 16 + row16], c);
    const int   cb = nt * 16;
    const float bs = b1[cb + row16];
    if (active) {
      #pragma unroll
      for (int r = 0; r < 8; ++r) {
        if (ebase + r + 8 * hi < n_edges) {
          float ev = c[r] + bs;
          ev = ev > 0.f ? ev : 0.f;              // relu(edge MLP)
          float v = nf[soff[r] + cb] + ev;       // + x_src (gather)
          v = v > 0.f ? v : 0.f;                 // relu(message)
          atomicAdd(&acc[doff[r] + cb], v);      // segment_sum (scatter)
        }
      }
    }
  }
}

// ---------------------------------------------------------------------------
// Output kernel: out = acc @ Wa + ba  (acc = node_feat + neigh, EPS = 0)
// One wave computes 16x128; A fragments register-resident (32 x v2f),
// packed Wa (64 KB) stays L2/WGP$-resident.
// ---------------------------------------------------------------------------
__global__ __launch_bounds__(256) void gine_out_kernel(
    const float* __restrict__ acc, const float* __restrict__ wap,
    const float* __restrict__ ba, float* __restrict__ out, int n_nodes)
{
  const int tid   = threadIdx.x;
  const int wave  = tid >> 5;
  const int lane  = tid & 31;
  const int row16 = lane & 15;
  const int hi    = lane >> 4;

  const int tile = blockIdx.x * WAVES + wave;
  const int m0   = tile * 16;
  if (m0 >= n_nodes) return;

  int arow = m0 + row16;
  if (arow >= n_nodes) arow = n_nodes - 1;
  const v2f* av    = (const v2f*)acc;          // 64 v2f per node row
  const int  abase = arow * (D_NODE / 2);

  v2f aa[D_NODE / 4];
  #pragma unroll
  for (int k = 0; k < D_NODE / 4; ++k) aa[k] = av[abase + 2 * k + hi];

  const v2f* wav = (const v2f*)wap;
  #pragma unroll 1
  for (int nt = 0; nt < D_NODE / 16; ++nt) {
    v8f c = {};
    #pragma unroll
    for (int k = 0; k < D_NODE / 4; ++k)
      c = wmma_f32(aa[k], wav[(2 * k + hi) * D_NODE + nt * 16 + row16], c);
    const int   col = nt * 16 + row16;
    const float bs  = ba[col];
    #pragma unroll
    for (int r = 0; r < 8; ++r) {
      int m = m0 + r + 8 * hi;
      if (m < n_nodes) out[m * D_NODE + col] = c[r] + bs;
    }
  }
}

extern "C" void kernel_launch(void* const* d_in, const int* in_sizes, int n_in,
                              void* d_out, int out_size, void* d_ws, size_t ws_size,
                              hipStream_t stream) {
  const float* nf = (const float*)d_in[0];
  const float* ef = (const float*)d_in[1];
  const float* W0 = (const float*)d_in[2];
  const float* b0 = (const float*)d_in[3];
  const float* W1 = (const float*)d_in[4];
  const float* b1 = (const float*)d_in[5];
  const float* Wa = (const float*)d_in[6];
  const float* ba = (const float*)d_in[7];
  const int*  src = (const int*)d_in[8];
  const int*  dst = (const int*)d_in[9];

  const int n_nodes = in_sizes[0] / D_NODE;
  const int n_edges = in_sizes[8];

  // Workspace layout (floats): acc | W0p | W1p | Wap
  float* acc = (float*)d_ws;
  float* w0p = acc + (size_t)n_nodes * D_NODE;
  float* w1p = w0p + D_EDGE * D_EDGE;
  float* wap = w1p + D_EDGE * D_NODE;

  // acc = node_feat (EPS = 0); scatter target for the edge kernel.
  hipMemcpyAsync(acc, nf, (size_t)n_nodes * D_NODE * sizeof(float),
                 hipMemcpyDeviceToDevice, stream);

  // Pre-pack weights K-pair-major (tiny, amortized over the whole layer).
  pack_pairs_kernel<<<(D_EDGE * D_EDGE + 255) / 256, 256, 0, stream>>>(
      W0, w0p, D_EDGE, D_EDGE);
  pack_pairs_kernel<<<(D_EDGE * D_NODE + 255) / 256, 256, 0, stream>>>(
      W1, w1p, D_EDGE, D_NODE);
  pack_pairs_kernel<<<(D_NODE * D_NODE + 255) / 256, 256, 0, stream>>>(
      Wa, wap, D_NODE, D_NODE);

  const int etiles  = (n_edges + 15) / 16;
  const int eblocks = (etiles + WAVES - 1) / WAVES;   // 6250 for 800k edges
  gine_edge_kernel<<<eblocks, 256, 0, stream>>>(ef, nf, w0p, b0, w1p, b1, src,
                                                dst, acc, n_edges);

  const int ntiles  = (n_nodes + 15) / 16;
  const int nblocks = (ntiles + WAVES - 1) / WAVES;   // 391 for 50k nodes
  gine_out_kernel<<<nblocks, 256, 0, stream>>>(acc, wap, ba, (float*)d_out,
                                               n_nodes);
}